// GGTLayer_46961172414536
// MI455X (gfx1250) — compile-verified
//
#include <hip/hip_runtime.h>
#include <hip/hip_bf16.h>

#define N_NODES 50000
#define N_EDGES 800000
#define D 64
#define EPSF 1e-12f
#define LN_EPSF 1e-5f

typedef float v2f __attribute__((ext_vector_type(2)));
typedef float v8f __attribute__((ext_vector_type(8)));

static __device__ __forceinline__ v8f wmma4(v2f a, v2f b, v8f c) {
  // D = A(16x4,f32) * B(4x16,f32) + C(16x16,f32)
  return __builtin_amdgcn_wmma_f32_16x16x4_f32(false, a, false, b, (short)0, c,
                                               false, false);
}

static __device__ __forceinline__ float sigmoidf_(float x) {
  return 1.0f / (1.0f + __expf(-x));
}

// Native fire-and-forget GLOBAL_ATOMIC_ADD_F32 (no CAS loop).
static __device__ __forceinline__ void atomic_add_f32(float* p, float v) {
  unsafeAtomicAdd(p, v);
}

// ---------------------------------------------------------------------------
// Kernel Z: zero the atomic accumulators (sum_sig, h_acc, p_acc)
// ---------------------------------------------------------------------------
__global__ __launch_bounds__(256) void zero_kernel(float4* __restrict__ ptr, int n4) {
  int i = blockIdx.x * blockDim.x + threadIdx.x;
  if (i < n4) ptr[i] = make_float4(0.f, 0.f, 0.f, 0.f);
}

// ---------------------------------------------------------------------------
// Kernel A: all node-side linears via V_WMMA_F32_16X16X4_F32.
// One wave per 16-node tile. A-frag layout (32b A 16x4): lane<16 -> row=lane,
// K={4k,4k+1}; lane>=16 -> row=lane-16, K={4k+2,4k+3}. B[k][n]=W[n][k] -> the
// B-frag is the contiguous float2 at W[n*K + 4k + 2*hi].
// ---------------------------------------------------------------------------
__global__ __launch_bounds__(256) void node_linear_kernel(
    const float* __restrict__ h, const float* __restrict__ p,
    const float* __restrict__ WK, const float* __restrict__ bK,
    const float* __restrict__ WV, const float* __restrict__ bV,
    const float* __restrict__ WB1, const float* __restrict__ bB1,
    const float* __restrict__ WB2, const float* __restrict__ bB2,
    const float* __restrict__ WC1, const float* __restrict__ bC1,
    const float* __restrict__ WC2, const float* __restrict__ bC2,
    float* __restrict__ sigmaQ, float* __restrict__ sigmaK,
    float* __restrict__ vh, float* __restrict__ B1h, float* __restrict__ B2h,
    float* __restrict__ C1p, float* __restrict__ C2p) {
  const int wave = (int)((blockIdx.x * blockDim.x + threadIdx.x) >> 5);
  if (wave >= N_NODES / 16) return;
  const int lane = threadIdx.x & 31;
  const int lo = lane & 15;
  const int hi = lane >> 4;
  const int row0 = wave * 16;

  // Load A fragments for h and p tiles (16x64 each) once.
  v2f ah[16], ap[16];
  const float* hrow = h + (size_t)(row0 + lo) * D + 2 * hi;
  const float* prow = p + (size_t)(row0 + lo) * D + 2 * hi;
#pragma unroll
  for (int ks = 0; ks < 16; ++ks) {
    ah[ks] = *(const v2f*)(hrow + ks * 4);
    ap[ks] = *(const v2f*)(prow + ks * 4);
  }

#pragma unroll
  for (int nt = 0; nt < 4; ++nt) {
    const int col = nt * 16 + lo;

    // ---- qk = [h,p] @ WK^T + bK  (K = 128) -> sigmaQ / sigmaK
    {
      v8f acc = 0.0f;
#pragma unroll
      for (int ks = 0; ks < 16; ++ks)
        acc = wmma4(ah[ks], *(const v2f*)(WK + (size_t)col * 128 + ks * 4 + 2 * hi), acc);
#pragma unroll
      for (int ks = 0; ks < 16; ++ks)
        acc = wmma4(ap[ks], *(const v2f*)(WK + (size_t)col * 128 + 64 + ks * 4 + 2 * hi), acc);
      const float bias = bK[col];
#pragma unroll
      for (int r = 0; r < 8; ++r) {
        const float q = acc[r] + bias;
        const size_t o = (size_t)(row0 + r + 8 * hi) * D + col;
        sigmaQ[o] = __expf(tanhf(q));
        sigmaK[o] = __expf(sigmoidf_(q));
      }
    }
    // ---- vh = [h,p] @ WV^T + bV
    {
      v8f acc = 0.0f;
#pragma unroll
      for (int ks = 0; ks < 16; ++ks)
        acc = wmma4(ah[ks], *(const v2f*)(WV + (size_t)col * 128 + ks * 4 + 2 * hi), acc);
#pragma unroll
      for (int ks = 0; ks < 16; ++ks)
        acc = wmma4(ap[ks], *(const v2f*)(WV + (size_t)col * 128 + 64 + ks * 4 + 2 * hi), acc);
      const float bias = bV[col];
#pragma unroll
      for (int r = 0; r < 8; ++r)
        vh[(size_t)(row0 + r + 8 * hi) * D + col] = acc[r] + bias;
    }
    // ---- B1h = h @ WB1^T + bB1
    {
      v8f acc = 0.0f;
#pragma unroll
      for (int ks = 0; ks < 16; ++ks)
        acc = wmma4(ah[ks], *(const v2f*)(WB1 + (size_t)col * D + ks * 4 + 2 * hi), acc);
      const float bias = bB1[col];
#pragma unroll
      for (int r = 0; r < 8; ++r)
        B1h[(size_t)(row0 + r + 8 * hi) * D + col] = acc[r] + bias;
    }
    // ---- B2h = h @ WB2^T + bB2
    {
      v8f acc = 0.0f;
#pragma unroll
      for (int ks = 0; ks < 16; ++ks)
        acc = wmma4(ah[ks], *(const v2f*)(WB2 + (size_t)col * D + ks * 4 + 2 * hi), acc);
      const float bias = bB2[col];
#pragma unroll
      for (int r = 0; r < 8; ++r)
        B2h[(size_t)(row0 + r + 8 * hi) * D + col] = acc[r] + bias;
    }
    // ---- C1p = p @ WC1^T + bC1
    {
      v8f acc = 0.0f;
#pragma unroll
      for (int ks = 0; ks < 16; ++ks)
        acc = wmma4(ap[ks], *(const v2f*)(WC1 + (size_t)col * D + ks * 4 + 2 * hi), acc);
      const float bias = bC1[col];
#pragma unroll
      for (int r = 0; r < 8; ++r)
        C1p[(size_t)(row0 + r + 8 * hi) * D + col] = acc[r] + bias;
    }
    // ---- C2p = p @ WC2^T + bC2
    {
      v8f acc = 0.0f;
#pragma unroll
      for (int ks = 0; ks < 16; ++ks)
        acc = wmma4(ap[ks], *(const v2f*)(WC2 + (size_t)col * D + ks * 4 + 2 * hi), acc);
      const float bias = bC2[col];
#pragma unroll
      for (int r = 0; r < 8; ++r)
        C2p[(size_t)(row0 + r + 8 * hi) * D + col] = acc[r] + bias;
    }
  }
}

// ---------------------------------------------------------------------------
// Kernel B: edge pass 1. One wave per 16-edge tile. WMMA B3e in registers
// (never hits HBM), then hat_eta = B3e + B1h[src] + B2h[dst], sigmoid ->
// atomic add into sum_sig[dst], store pre-ReLU hat_eta into e_out, and reduce
// alpha = <sigmaQ[src], sigmaK[dst]> with 16-lane shuffles.
// ---------------------------------------------------------------------------
__global__ __launch_bounds__(256) void edge_pass1_kernel(
    const float* __restrict__ e, const float* __restrict__ WB3,
    const float* __restrict__ bB3,
    const int* __restrict__ src, const int* __restrict__ dst,
    const float* __restrict__ B1h, const float* __restrict__ B2h,
    const float* __restrict__ sigmaQ, const float* __restrict__ sigmaK,
    float* __restrict__ sum_sig, float* __restrict__ alpha,
    float* __restrict__ e_out) {
  const int wave = (int)((blockIdx.x * blockDim.x + threadIdx.x) >> 5);
  if (wave >= N_EDGES / 16) return;
  const int lane = threadIdx.x & 31;
  const int lo = lane & 15;
  const int hi = lane >> 4;
  const int base = wave * 16;

  v2f ae[16];
  const float* erow = e + (size_t)(base + lo) * D + 2 * hi;
#pragma unroll
  for (int ks = 0; ks < 16; ++ks) ae[ks] = *(const v2f*)(erow + ks * 4);

  v8f acc[4];
#pragma unroll
  for (int nt = 0; nt < 4; ++nt) {
    v8f a = 0.0f;
    const int col = nt * 16 + lo;
#pragma unroll
    for (int ks = 0; ks < 16; ++ks)
      a = wmma4(ae[ks], *(const v2f*)(WB3 + (size_t)col * D + ks * 4 + 2 * hi), a);
    acc[nt] = a;
  }

  float b3[4];
#pragma unroll
  for (int nt = 0; nt < 4; ++nt) b3[nt] = bB3[nt * 16 + lo];

#pragma unroll
  for (int r = 0; r < 8; ++r) {
    const int ei = base + r + 8 * hi;   // edge for this half-wave & r
    const int s = src[ei];
    const int d0 = dst[ei];
    float pr = 0.0f;
#pragma unroll
    for (int nt = 0; nt < 4; ++nt) {
      const int f = nt * 16 + lo;
      const float hat = acc[nt][r] + b3[nt] +
                        B1h[(size_t)s * D + f] + B2h[(size_t)d0 * D + f];
      const float sg = sigmoidf_(hat);
      atomic_add_f32(&sum_sig[(size_t)d0 * D + f], sg);
      e_out[(size_t)ei * D + f] = hat;  // pre-ReLU; pass 2 applies ReLU
      pr += sigmaQ[(size_t)s * D + f] * sigmaK[(size_t)d0 * D + f];
    }
    pr += __shfl_xor(pr, 8, 16);
    pr += __shfl_xor(pr, 4, 16);
    pr += __shfl_xor(pr, 2, 16);
    pr += __shfl_xor(pr, 1, 16);
    if (lo == 0) alpha[ei] = pr;
  }
}

// ---------------------------------------------------------------------------
// Kernel C: edge pass 2 (elementwise over E x D). eta = sig/(sum_sig+eps),
// scatter-add into h_acc / p_acc, ReLU e_out in place.
// ---------------------------------------------------------------------------
__global__ __launch_bounds__(256) void edge_pass2_kernel(
    const int* __restrict__ src, const int* __restrict__ dst,
    const float* __restrict__ alpha, const float* __restrict__ sum_sig,
    const float* __restrict__ vh, const float* __restrict__ C2p,
    float* __restrict__ h_acc, float* __restrict__ p_acc,
    float* __restrict__ e_out) {
  const unsigned idx = blockIdx.x * blockDim.x + threadIdx.x;
  if (idx >= (unsigned)N_EDGES * D) return;
  const int ei = (int)(idx >> 6);
  const int f = (int)(idx & 63);
  const int s = src[ei];
  const int d0 = dst[ei];
  const float hat = e_out[idx];
  const float sg = sigmoidf_(hat);
  const float eta = sg / (sum_sig[(size_t)d0 * D + f] + EPSF);
  const float al = alpha[ei];
  atomic_add_f32(&h_acc[(size_t)d0 * D + f], eta * al * vh[(size_t)s * D + f]);
  atomic_add_f32(&p_acc[(size_t)d0 * D + f], eta * C2p[(size_t)s * D + f]);
  e_out[idx] = fmaxf(hat, 0.0f);
}

// ---------------------------------------------------------------------------
// Kernel D: node finalize. One wave per node: ReLU + LayerNorm (shuffle
// reduction over 64 features, 2 per lane) for h_out; tanh for p_out.
// ---------------------------------------------------------------------------
__global__ __launch_bounds__(256) void node_final_kernel(
    const float* __restrict__ vh, const float* __restrict__ h_acc,
    const float* __restrict__ C1p, const float* __restrict__ p_acc,
    const float* __restrict__ ln_g, const float* __restrict__ ln_b,
    float* __restrict__ h_out, float* __restrict__ p_out) {
  const int node = (int)((blockIdx.x * blockDim.x + threadIdx.x) >> 5);
  if (node >= N_NODES) return;
  const int lane = threadIdx.x & 31;
  const int f0 = lane;
  const int f1 = lane + 32;
  const size_t o0 = (size_t)node * D + f0;
  const size_t o1 = (size_t)node * D + f1;
  const float x0 = fmaxf(vh[o0] + h_acc[o0], 0.0f);
  const float x1 = fmaxf(vh[o1] + h_acc[o1], 0.0f);
  float s = x0 + x1;
  float ss = x0 * x0 + x1 * x1;
#pragma unroll
  for (int m = 16; m >= 1; m >>= 1) {
    s += __shfl_xor(s, m, 32);
    ss += __shfl_xor(ss, m, 32);
  }
  const float mean = s * (1.0f / 64.0f);
  const float var = ss * (1.0f / 64.0f) - mean * mean;
  const float inv = rsqrtf(var + LN_EPSF);
  h_out[o0] = (x0 - mean) * inv * ln_g[f0] + ln_b[f0];
  h_out[o1] = (x1 - mean) * inv * ln_g[f1] + ln_b[f1];
  p_out[o0] = tanhf(C1p[o0] + p_acc[o0]);
  p_out[o1] = tanhf(C1p[o1] + p_acc[o1]);
}

// ---------------------------------------------------------------------------
extern "C" void kernel_launch(void* const* d_in, const int* in_sizes, int n_in,
                              void* d_out, int out_size, void* d_ws, size_t ws_size,
                              hipStream_t stream) {
  (void)in_sizes; (void)n_in; (void)out_size; (void)ws_size;
  const float* h   = (const float*)d_in[0];
  const float* e   = (const float*)d_in[1];
  const float* p   = (const float*)d_in[2];
  const float* WK  = (const float*)d_in[3];
  const float* bK  = (const float*)d_in[4];
  const float* WV  = (const float*)d_in[5];
  const float* bV  = (const float*)d_in[6];
  const float* WB1 = (const float*)d_in[7];
  const float* bB1 = (const float*)d_in[8];
  const float* WB2 = (const float*)d_in[9];
  const float* bB2 = (const float*)d_in[10];
  const float* WB3 = (const float*)d_in[11];
  const float* bB3 = (const float*)d_in[12];
  const float* WC1 = (const float*)d_in[13];
  const float* bC1 = (const float*)d_in[14];
  const float* WC2 = (const float*)d_in[15];
  const float* bC2 = (const float*)d_in[16];
  const float* ln_g = (const float*)d_in[17];
  const float* ln_b = (const float*)d_in[18];
  const int* src = (const int*)d_in[19];
  const int* dst = (const int*)d_in[20];

  const size_t ND = (size_t)N_NODES * D;
  float* ws = (float*)d_ws;
  float* sigmaQ  = ws + 0 * ND;
  float* sigmaK  = ws + 1 * ND;
  float* vh      = ws + 2 * ND;
  float* B1h     = ws + 3 * ND;
  float* B2h     = ws + 4 * ND;
  float* C1p     = ws + 5 * ND;
  float* C2p     = ws + 6 * ND;
  float* sum_sig = ws + 7 * ND;   // |
  float* h_acc   = ws + 8 * ND;   // | contiguous: zeroed together
  float* p_acc   = ws + 9 * ND;   // |
  float* alpha   = ws + 10 * ND;  // N_EDGES floats

  float* h_out = (float*)d_out;
  float* e_out = h_out + ND;
  float* p_out = e_out + (size_t)N_EDGES * D;

  // 1) zero accumulators (3*ND floats, ND%4==0)
  {
    const int n4 = (int)(3 * ND / 4);
    zero_kernel<<<(n4 + 255) / 256, 256, 0, stream>>>((float4*)sum_sig, n4);
  }
  // 2) node linears (WMMA): one wave per 16-node tile
  {
    const int waves = N_NODES / 16;  // 3125
    const int blocks = (waves * 32 + 255) / 256;
    node_linear_kernel<<<blocks, 256, 0, stream>>>(
        h, p, WK, bK, WV, bV, WB1, bB1, WB2, bB2, WC1, bC1, WC2, bC2,
        sigmaQ, sigmaK, vh, B1h, B2h, C1p, C2p);
  }
  // 3) edge pass 1 (WMMA B3e fused with gather/sigmoid/segment-sum/alpha)
  {
    const int waves = N_EDGES / 16;  // 50000
    const int blocks = (waves * 32 + 255) / 256;
    edge_pass1_kernel<<<blocks, 256, 0, stream>>>(
        e, WB3, bB3, src, dst, B1h, B2h, sigmaQ, sigmaK,
        sum_sig, alpha, e_out);
  }
  // 4) edge pass 2 (eta, scatter-adds, ReLU e_out)
  {
    const unsigned total = (unsigned)N_EDGES * D;
    edge_pass2_kernel<<<(total + 255) / 256, 256, 0, stream>>>(
        src, dst, alpha, sum_sig, vh, C2p, h_acc, p_acc, e_out);
  }
  // 5) node finalize (ReLU+LayerNorm, tanh)
  {
    const int blocks = (N_NODES * 32 + 255) / 256;
    node_final_kernel<<<blocks, 256, 0, stream>>>(
        vh, h_acc, C1p, p_acc, ln_g, ln_b, h_out, p_out);
  }
}